// GraphConv_47974784696629
// MI455X (gfx1250) — compile-verified
//
#include <hip/hip_runtime.h>
#include <hip/hip_bf16.h>

// ---------------------------------------------------------------------------
// GATv2 4-layer GNN forward for MI455X (gfx1250, wave32).
// Node projection GEMMs use V_WMMA_F32_16X16X4_F32 with a zero-padded,
// transposed weight staging buffer so the K-loop is branch-free (no exec
// divergence, b64 loads). Edge phase is L2-resident gather/scatter with the
// 16xHC edge projection staged in LDS.
// ---------------------------------------------------------------------------

typedef float v2f __attribute__((ext_vector_type(2)));
typedef float v8f __attribute__((ext_vector_type(8)));

#define N_NODES   50000
#define N_EDGES   800000
#define E_TOT     850000   // N_EDGES + N_NODES self loops
#define EDGE_F    16
#define NUM_GRAPHS 64
#define NEG_SLOPE 0.2f

// ---- monotone float<->uint key for atomicMax-based segment max -------------
__device__ __forceinline__ unsigned fkey(float f) {
  unsigned b = __float_as_uint(f);
  return (b & 0x80000000u) ? ~b : (b | 0x80000000u);
}
__device__ __forceinline__ float funkey(unsigned k) {
  unsigned b = (k & 0x80000000u) ? (k & 0x7fffffffu) : ~k;
  return __uint_as_float(b);
}

// ---- utility fills ---------------------------------------------------------
__global__ void fill_f32(float* p, float v, long n) {
  long i = (long)blockIdx.x * blockDim.x + threadIdx.x;
  if (i < n) p[i] = v;
}
__global__ void fill_u32(unsigned* p, unsigned v, long n) {
  long i = (long)blockIdx.x * blockDim.x + threadIdx.x;
  if (i < n) p[i] = v;
}

// ---- self-loop edge_attr: mean of incoming edge_attr per node --------------
__global__ void loop_attr_accum(const int* __restrict__ ei,
                                const float* __restrict__ eattr,
                                float* __restrict__ loop,
                                float* __restrict__ deg) {
  int e = blockIdx.x * blockDim.x + threadIdx.x;
  if (e >= N_EDGES) return;
  int dst = ei[N_EDGES + e];
  atomicAdd(&deg[dst], 1.0f);
  const float* ea = eattr + (size_t)e * EDGE_F;
  float* la = loop + (size_t)dst * EDGE_F;
#pragma unroll
  for (int k = 0; k < EDGE_F; ++k) atomicAdd(&la[k], ea[k]);
}
__global__ void loop_attr_final(float* __restrict__ loop,
                                const float* __restrict__ deg) {
  int i = blockIdx.x * blockDim.x + threadIdx.x;
  if (i >= N_NODES * EDGE_F) return;
  float d = deg[i / EDGE_F];
  loop[i] = (d > 0.f) ? loop[i] / d : 0.f;
}

// ---- stage weights: WpT[n*Kpad + k] = (n<HC && k<K) ? W[k*HC+n] : 0 --------
// Transposed + zero padded: makes the GEMM K-loop fully branch-free, and each
// lane's (k, k+1) B-pair contiguous & 8B-aligned (Kpad % 8 == 0).
__global__ void pad_weightT(const float* __restrict__ W, float* __restrict__ WpT,
                            int K, int Kpad, int HC, int NP) {
  int i = blockIdx.x * blockDim.x + threadIdx.x;
  if (i >= NP * Kpad) return;
  int n = i / Kpad, k = i - n * Kpad;
  WpT[i] = (n < HC && k < K) ? W[(size_t)k * HC + n] : 0.f;
}

// ---- WMMA fp32 GEMM: out[N_NODES x HC] = A[N_NODES x K] * W[K x HC] + bias -
// One wave per 16x16 C tile; V_WMMA_F32_16X16X4_F32, K stepped by 8 (2 WMMA /
// iter). Zero-padded B kills any A over-read tail product, so no bounds
// checks inside the loop (EXEC stays all-1s for WMMA).
__global__ __launch_bounds__(256)
void gemm_wmma(const float* __restrict__ A, const float* __restrict__ WpT,
               const float* __restrict__ bias, float* __restrict__ out,
               int K, int Kpad, int HC, int ntN, int ntiles) {
  int wave = blockIdx.x * (blockDim.x >> 5) + (threadIdx.x >> 5);
  if (wave >= ntiles) return;                 // wave-uniform exit
  int tm = wave / ntN, tn = wave - tm * ntN;
  int row0 = tm << 4, col0 = tn << 4;
  int lane = threadIdx.x & 31;
  int lm = lane & 15, lh = lane >> 4;
  int col = col0 + lm;                        // always < NP (valid in WpT)

  v8f acc = {0.f, 0.f, 0.f, 0.f, 0.f, 0.f, 0.f, 0.f};
  const float* arow = A + (size_t)(row0 + lm) * K + (lh << 1);
  const float* wcol = WpT + (size_t)col * Kpad + (lh << 1);
  for (int k0 = 0; k0 < Kpad; k0 += 8) {
    v2f a0, a1;
    a0.x = arow[k0];     a0.y = arow[k0 + 1];
    a1.x = arow[k0 + 4]; a1.y = arow[k0 + 5];
    v2f b0 = *(const v2f*)(wcol + k0);        // 8B-aligned b64 load
    v2f b1 = *(const v2f*)(wcol + k0 + 4);
    __builtin_prefetch(arow + k0 + 64, 0, 1); // global_prefetch_b8 (A stream)
    acc = __builtin_amdgcn_wmma_f32_16x16x4_f32(false, a0, false, b0,
                                                (short)0, acc, false, false);
    acc = __builtin_amdgcn_wmma_f32_16x16x4_f32(false, a1, false, b1,
                                                (short)0, acc, false, false);
  }
  if (col < HC) {
    float bv = bias[col];
    int rbase = row0 + (lh << 3);             // D: vgpr r -> row 8*lh + r
#pragma unroll
    for (int r = 0; r < 8; ++r)
      out[(size_t)(rbase + r) * HC + col] = acc[r] + bv;
  }
}

// ---- per-(edge,head) attention logit + running segment max -----------------
// We (16 x HC) staged transposed in LDS -> the 16-FMA inner product reads 16
// contiguous LDS floats (ds_load_b128) instead of 16 strided VMEM loads.
__global__ __launch_bounds__(256)
void edge_logits(const float* __restrict__ xl,
                 const float* __restrict__ xr,
                 const int* __restrict__ ei,
                 const float* __restrict__ eattr,
                 const float* __restrict__ loop,
                 const float* __restrict__ We,
                 const float* __restrict__ att,
                 float* __restrict__ logit,
                 unsigned* __restrict__ amax,
                 int H, int C, int HC) {
  __shared__ float sWeT[300 * EDGE_F];        // [hc][k], 19.2 KB max
  __shared__ float sAtt[300];
  for (int i = threadIdx.x; i < HC * EDGE_F; i += blockDim.x) {
    int hc = i >> 4, k = i & 15;
    sWeT[i] = We[(size_t)k * HC + hc];
  }
  for (int i = threadIdx.x; i < HC; i += blockDim.x) sAtt[i] = att[i];
  __syncthreads();

  int t = blockIdx.x * blockDim.x + threadIdx.x;
  if (t >= E_TOT * H) return;
  int e = t / H, h = t - e * H;
  int src, dst;
  const float* ea;
  if (e < N_EDGES) {
    src = ei[e]; dst = ei[N_EDGES + e];
    ea = eattr + (size_t)e * EDGE_F;
  } else {
    src = dst = e - N_EDGES;
    ea = loop + (size_t)src * EDGE_F;
  }
  float eav[EDGE_F];
#pragma unroll
  for (int k = 0; k < EDGE_F; ++k) eav[k] = ea[k];

  const float* xls = xl + (size_t)src * HC + h * C;
  const float* xrd = xr + (size_t)dst * HC + h * C;
  const float* av  = sAtt + h * C;
  const float* wb  = sWeT + (size_t)(h * C) * EDGE_F;
  float acc = 0.f;
  for (int c = 0; c < C; ++c) {
    float s = xls[c] + xrd[c];
    const float* wc = wb + c * EDGE_F;
#pragma unroll
    for (int k = 0; k < EDGE_F; ++k) s += eav[k] * wc[k];
    s = (s > 0.f) ? s : NEG_SLOPE * s;        // LeakyReLU(0.2)
    acc += s * av[c];
  }
  logit[t] = acc;
  atomicMax(&amax[(size_t)dst * H + h], fkey(acc));
}

// ---- w = exp(logit - amax[dst]); denom[dst] += w ---------------------------
__global__ void edge_softmax_w(const int* __restrict__ ei,
                               float* __restrict__ logit,
                               const unsigned* __restrict__ amax,
                               float* __restrict__ denom, int H) {
  int t = blockIdx.x * blockDim.x + threadIdx.x;
  if (t >= E_TOT * H) return;
  int e = t / H, h = t - e * H;
  int dst = (e < N_EDGES) ? ei[N_EDGES + e] : e - N_EDGES;
  float m = funkey(amax[(size_t)dst * H + h]);
  float w = __expf(logit[t] - m);
  logit[t] = w;
  atomicAdd(&denom[(size_t)dst * H + h], w);
}

// ---- out[dst] += alpha * xl[src] ------------------------------------------
__global__ void edge_aggregate(const int* __restrict__ ei,
                               const float* __restrict__ xl,
                               const float* __restrict__ logit,
                               const float* __restrict__ denom,
                               float* __restrict__ out,
                               int H, int C, int HC) {
  int t = blockIdx.x * blockDim.x + threadIdx.x;
  if (t >= E_TOT * H) return;
  int e = t / H, h = t - e * H;
  int src, dst;
  if (e < N_EDGES) { src = ei[e]; dst = ei[N_EDGES + e]; }
  else             { src = dst = e - N_EDGES; }
  float alpha = logit[t] / (denom[(size_t)dst * H + h] + 1e-16f);
  const float* xs = xl + (size_t)src * HC + h * C;
  float* od = out + (size_t)dst * HC + h * C;
  for (int c = 0; c < C; ++c) atomicAdd(&od[c], xs[c] * alpha);
}

// ---- bias add (+ optional ReLU) -------------------------------------------
__global__ void bias_act(float* __restrict__ out, const float* __restrict__ bias,
                         int HC, int relu, long n) {
  long i = (long)blockIdx.x * blockDim.x + threadIdx.x;
  if (i >= n) return;
  float v = out[i] + bias[(int)(i % HC)];
  if (relu) v = (v > 0.f) ? v : 0.f;
  out[i] = v;
}

// ---- global max pool (h >= 0 post-ReLU -> raw-bit atomicMax is ordered) ----
__global__ void pool_max(const float* __restrict__ h, const int* __restrict__ batch,
                         unsigned* __restrict__ pool, int HC) {
  long i = (long)blockIdx.x * blockDim.x + threadIdx.x;
  long n = (long)N_NODES * HC;
  if (i >= n) return;
  int node = (int)(i / HC);
  int c = (int)(i - (long)node * HC);
  atomicMax(&pool[(size_t)batch[node] * HC + c], __float_as_uint(h[i]));
}

// ---- head MLP: relu(g@fc1+b1)@fc2+b2 --------------------------------------
__global__ void head_fc(const float* __restrict__ pool,
                        const float* __restrict__ w1, const float* __restrict__ b1,
                        const float* __restrict__ w2, const float* __restrict__ b2,
                        float* __restrict__ out) {
  int g = threadIdx.x;
  if (g >= NUM_GRAPHS) return;
  const float* pg = pool + g * 300;
  float o = b2[0];
  for (int j = 0; j < 10; ++j) {
    float s = b1[j];
    for (int c = 0; c < 300; ++c) s += pg[c] * w1[c * 10 + j];
    s = (s > 0.f) ? s : 0.f;
    o += s * w2[j];
  }
  out[g] = o;
}

// ---------------------------------------------------------------------------
extern "C" void kernel_launch(void* const* d_in, const int* in_sizes, int n_in,
                              void* d_out, int out_size, void* d_ws, size_t ws_size,
                              hipStream_t stream) {
  const float* x     = (const float*)d_in[0];
  const int*   ei    = (const int*)d_in[1];
  const float* eattr = (const float*)d_in[2];
  const int*   batch = (const int*)d_in[3];
  // params flattened in setup_inputs() insertion order:
  // per layer l: [4+7l] Wl, bl, Wr, br, We, att, bias ; then fc1_w,b, fc2_w,b

  float* ws = (float*)d_ws;
  float*    A    = ws;                        // 15,000,000 f (node buf, in/out)
  float*    XL   = ws + 15000000ull;          // 15,000,000 f (x @ Wl)
  float*    XR   = ws + 30000000ull;          // 15,000,000 f (x @ Wr)
  float*    LOOP = ws + 45000000ull;          //    800,000 f (self-loop attr)
  float*    DEG  = ws + 45800000ull;          //     50,000 f
  float*    LOG  = ws + 45850000ull;          //  2,550,000 f (logits -> w)
  unsigned* AMAX = (unsigned*)(ws + 48400000ull); // 150,000 u32
  float*    DEN  = ws + 48550000ull;          //    150,000 f
  unsigned* POOL = (unsigned*)(ws + 48700000ull); // 19,200 u32 (float bits)
  float*    WPT  = ws + 48720000ull;          //   <=92,416 f (padded W^T)

  const int TB = 256;
  auto nb = [](long n) { return (int)((n + 255) / 256); };

  // self-loop attr = mean of incoming edge_attr (LOOP and DEG are contiguous)
  fill_f32<<<nb(850000), TB, 0, stream>>>(LOOP, 0.f, 850000);
  loop_attr_accum<<<nb(N_EDGES), TB, 0, stream>>>(ei, eattr, LOOP, DEG);
  loop_attr_final<<<nb((long)N_NODES * EDGE_F), TB, 0, stream>>>(LOOP, DEG);

  const int LDIN[4] = {128, 75, 150, 300};
  const int LC[4]   = {25, 50, 100, 300};
  const int LH[4]   = {3, 3, 3, 1};

  const float* cur = x;
  for (int l = 0; l < 4; ++l) {
    int din = LDIN[l], C = LC[l], H = LH[l], HC = H * C;
    const float* Wl   = (const float*)d_in[4 + 7 * l + 0];
    const float* bl   = (const float*)d_in[4 + 7 * l + 1];
    const float* Wr   = (const float*)d_in[4 + 7 * l + 2];
    const float* br   = (const float*)d_in[4 + 7 * l + 3];
    const float* We   = (const float*)d_in[4 + 7 * l + 4];
    const float* att  = (const float*)d_in[4 + 7 * l + 5];
    const float* bias = (const float*)d_in[4 + 7 * l + 6];

    int Kpad = (din + 7) & ~7;
    int ntN = (HC + 15) >> 4;
    int NP = ntN << 4;
    int ntiles = (N_NODES / 16) * ntN;
    int gblocks = (ntiles + 7) / 8;           // 8 waves (tiles) per block

    pad_weightT<<<nb((long)NP * Kpad), TB, 0, stream>>>(Wl, WPT, din, Kpad, HC, NP);
    gemm_wmma<<<gblocks, 256, 0, stream>>>(cur, WPT, bl, XL, din, Kpad, HC, ntN, ntiles);
    pad_weightT<<<nb((long)NP * Kpad), TB, 0, stream>>>(Wr, WPT, din, Kpad, HC, NP);
    gemm_wmma<<<gblocks, 256, 0, stream>>>(cur, WPT, br, XR, din, Kpad, HC, ntN, ntiles);

    fill_u32<<<nb((long)N_NODES * H), TB, 0, stream>>>(AMAX, 0u, (long)N_NODES * H);
    fill_f32<<<nb((long)N_NODES * H), TB, 0, stream>>>(DEN, 0.f, (long)N_NODES * H);

    long th = (long)E_TOT * H;
    edge_logits<<<nb(th), TB, 0, stream>>>(XL, XR, ei, eattr, LOOP, We, att,
                                           LOG, AMAX, H, C, HC);
    edge_softmax_w<<<nb(th), TB, 0, stream>>>(ei, LOG, AMAX, DEN, H);

    // input buffer `cur` (== A for l>0) is dead after the GEMMs -> reuse as out
    fill_f32<<<nb((long)N_NODES * HC), TB, 0, stream>>>(A, 0.f, (long)N_NODES * HC);
    edge_aggregate<<<nb(th), TB, 0, stream>>>(ei, XL, LOG, DEN, A, H, C, HC);

    int relu = (l == 1 || l == 3);
    bias_act<<<nb((long)N_NODES * HC), TB, 0, stream>>>(A, bias, HC, relu,
                                                        (long)N_NODES * HC);
    cur = A;
  }

  fill_u32<<<nb((long)NUM_GRAPHS * 300), TB, 0, stream>>>(POOL, 0u,
                                                          (long)NUM_GRAPHS * 300);
  pool_max<<<nb((long)N_NODES * 300), TB, 0, stream>>>(A, batch, POOL, 300);
  head_fc<<<1, 64, 0, stream>>>((const float*)POOL,
                                (const float*)d_in[32], (const float*)d_in[33],
                                (const float*)d_in[34], (const float*)d_in[35],
                                (float*)d_out);
}